// LengthRegulator_57878979281201
// MI455X (gfx1250) — compile-verified
//
#include <hip/hip_runtime.h>
#include <stdint.h>

typedef float __attribute__((ext_vector_type(4))) f32x4;
typedef unsigned int u32;
typedef unsigned int __attribute__((ext_vector_type(4))) u32x4;
typedef int __attribute__((ext_vector_type(4))) i32x4;
typedef int __attribute__((ext_vector_type(8))) i32x8;

#define B_  16
#define TX  512
#define TY  2048
#define DD  256

// TDM zero-fill tiling: 16 KB LDS tile, 16 iterations -> 256 KB per block.
#define TDM_TILE_BYTES 16384
#define TDM_ITER       16
#define TDM_BLOCK_BYTES (TDM_TILE_BYTES * TDM_ITER)     // 262144
#define ATTN_BYTES ((size_t)B_ * TX * TY * 4)           // 64 MB
#define TDM_NBLOCKS ((int)(ATTN_BYTES / TDM_BLOCK_BYTES)) // 256

#if defined(__has_builtin)
#if __has_builtin(__builtin_amdgcn_tensor_store_from_lds)
#define HAVE_TDM 1
#endif
#endif

// ---------------------------------------------------------------------------
// Kernel 1a (CDNA5 TDM path): zero the 64 MB attn region via the Tensor Data
// Mover. Each block zeroes one 16 KB LDS tile, then wave 0 issues a single
// tensor_store_from_lds with iterate_enable=1 (lds_addr_increment=0,
// global_addr_increment=16 KB, 16 iterations) -> 256 KB streamed per block
// by the DMA engine, tracked with TENSORcnt.
// ---------------------------------------------------------------------------
#ifdef HAVE_TDM
__global__ __launch_bounds__(256) void zero_attn_tdm_kernel(float* __restrict__ attn) {
    __shared__ float s_zero[TDM_TILE_BYTES / 4];   // first/only LDS object -> offset 0
    for (int i = threadIdx.x; i < TDM_TILE_BYTES / 4; i += 256) s_zero[i] = 0.f;
    __syncthreads();                               // DS stores complete before TDM reads LDS

    if (threadIdx.x == 0) {                        // one TDM op per block (EXEC ignored by TDM)
        const unsigned long long gaddr =
            (unsigned long long)(uintptr_t)attn +
            (unsigned long long)blockIdx.x * (unsigned long long)TDM_BLOCK_BYTES;

        // ----- D# group 0 (128b): count | lds_addr | global_addr[56:0] | type=2
        u32x4 g0 = {
            1u,                                          // count=1 (valid), is_restore=0, gather=0
            0u,                                          // lds_addr = 0 (s_zero at LDS base)
            (u32)(gaddr & 0xFFFFFFFFu),                  // global_addr[31:0]
            (u32)((gaddr >> 32) & 0x01FFFFFFu) | (2u << 30)  // global_addr[56:32] | type=2
        };

        // ----- D# group 1 (256b)
        // word0: data_size=3 (8B) in [17:16], iterate_enable in [19]
        // tensor_dim0 = 0x7FFFFFFF (never clips), tensor_dim1 = 1
        // tile_dim0 = 2048 elements (x8B = 16 KB), tile_dim1/2 = 0 (1D tile)
        // tensor_dim0_stride = 2048
        i32x8 g1 = {
            (3 << 16) | (1 << 19),        // word0
            (int)0xFFFF0000,              // word1: tensor_dim0[15:0]=0xFFFF in [63:48]
            0x00017FFF,                   // word2: tensor_dim0[31:16]=0x7FFF, tensor_dim1[15:0]=1
            (2048 << 16),                 // word3: tensor_dim1[31:16]=0, tile_dim0=2048
            0,                            // word4: tile_dim1=0, tile_dim2=0
            2048,                         // word5: tensor_dim0_stride[31:0]
            0,                            // word6
            0                             // word7
        };

        // ----- D# group 2 (128b), iterate mode:
        // tensor_dim2=1 | lds_addr_increment=0 | global_addr_increment=2048 elems (16 KB)
        // | iterate_count = TDM_ITER-1 (0 => 1x)
        i32x4 g2 = { 1, 0, 2048, (TDM_ITER - 1) << 16 };
        i32x4 g3 = { 0, 0, 0, 0 };

#if __clang_major__ >= 23
        i32x8 g4 = { 0, 0, 0, 0, 0, 0, 0, 0 };
        __builtin_amdgcn_tensor_store_from_lds(g0, g1, g2, g3, g4, 0);
#else
        __builtin_amdgcn_tensor_store_from_lds(g0, g1, g2, g3, 0);
#endif
        __builtin_amdgcn_s_wait_tensorcnt(0);      // keep LDS alive until DMA drained
    }
}
#endif

// ---------------------------------------------------------------------------
// Kernel 1b (fallback): plain b128 grid-stride zero-fill.
// ---------------------------------------------------------------------------
__global__ __launch_bounds__(256) void zero_attn_kernel(f32x4* __restrict__ p, int n4) {
    int i      = blockIdx.x * blockDim.x + threadIdx.x;
    int stride = gridDim.x * blockDim.x;
    f32x4 z = {0.f, 0.f, 0.f, 0.f};
    for (; i < n4; i += stride) p[i] = z;
}

// ---------------------------------------------------------------------------
// Kernel 2: per-batch inclusive scan of durations (Tx=512) + inverse map
// ty -> tx, all in LDS (10 KB of 320 KB/WGP). One 512-thread block per batch.
// ---------------------------------------------------------------------------
__global__ __launch_bounds__(512) void scan_map_kernel(const int* __restrict__ durations,
                                                       int* __restrict__ txmap) {
    __shared__ int s_cum[TX];
    __shared__ int s_map[TY];

    const int b = blockIdx.x;
    const int t = threadIdx.x;

    const int d = durations[b * TX + t];
    s_cum[t] = d;
    __syncthreads();

    #pragma unroll
    for (int off = 1; off < TX; off <<= 1) {       // Hillis-Steele, 9 steps
        int v = (t >= off) ? s_cum[t - off] : 0;
        __syncthreads();
        s_cum[t] += v;
        __syncthreads();
    }

    for (int i = t; i < TY; i += TX) s_map[i] = -1;
    __syncthreads();

    const int hi = s_cum[t];
    const int lo = hi - d;                          // cum[t-1]
    for (int j = lo; j < hi && j < TY; ++j) s_map[j] = t;
    __syncthreads();

    for (int i = t; i < TY; i += TX) txmap[b * TY + i] = s_map[i];
}

// ---------------------------------------------------------------------------
// Kernel 3: one wave32 per (b, ty) output row.
//   mu_y[b,ty,:] = m * x[b,tx,:]   (m = x_mask*y_mask; zeros if unmapped)
//   attn[b,tx,ty] = m              (one element per mapped row)
// mu_y is write-once/never-read -> nontemporal stores, preserving L2 for attn.
// ---------------------------------------------------------------------------
__global__ __launch_bounds__(256) void gather_scatter_kernel(
        const float* __restrict__ x,
        const float* __restrict__ x_mask,
        const float* __restrict__ y_mask,
        const int*   __restrict__ txmap,
        float* __restrict__ mu_y,
        float* __restrict__ attn) {
    const int lane = threadIdx.x & 31;            // wave32 lane
    const int wid  = threadIdx.x >> 5;            // 8 waves per block
    const int row  = blockIdx.x * 8 + wid;        // row = b*TY + ty
    const int b    = row >> 11;                   // TY = 2048
    const int ty   = row & (TY - 1);

    const int tx = txmap[row];
    f32x4* __restrict__ dst = reinterpret_cast<f32x4*>(mu_y) + (size_t)row * (DD / 4);

    if (tx >= 0) {
        const float m = x_mask[b * TX + tx] * y_mask[row];
        const f32x4* __restrict__ src =
            reinterpret_cast<const f32x4*>(x + ((size_t)b * TX + tx) * DD);
        __builtin_prefetch(src + lane + 32, 0, 0);   // gfx1250 global_prefetch_b8
        f32x4 v0 = src[lane];
        f32x4 v1 = src[lane + 32];
        v0 *= m;
        v1 *= m;
        __builtin_nontemporal_store(v0, dst + lane);
        __builtin_nontemporal_store(v1, dst + lane + 32);
        if (lane == 0) attn[((size_t)b * TX + tx) * TY + ty] = m;
    } else {
        f32x4 z = {0.f, 0.f, 0.f, 0.f};
        __builtin_nontemporal_store(z, dst + lane);
        __builtin_nontemporal_store(z, dst + lane + 32);
    }
}

// ---------------------------------------------------------------------------
// Host-side launcher. Inputs (setup_inputs order):
//   0:x 1:x_lengths 2:x_mask 3:y 4:y_lengths 5:y_mask 6:logw 7:durations(int)
// d_out = [mu_y (B*TY*D floats) | attn (B*TX*TY floats)]
// d_ws  = txmap (B*TY ints = 128 KB)
// ---------------------------------------------------------------------------
extern "C" void kernel_launch(void* const* d_in, const int* in_sizes, int n_in,
                              void* d_out, int out_size, void* d_ws, size_t ws_size,
                              hipStream_t stream) {
    (void)in_sizes; (void)n_in; (void)out_size; (void)ws_size;

    const float* x         = (const float*)d_in[0];
    const float* x_mask    = (const float*)d_in[2];
    const float* y_mask    = (const float*)d_in[5];
    const int*   durations = (const int*)d_in[7];

    float* mu_y = (float*)d_out;
    float* attn = (float*)d_out + (size_t)B_ * TY * DD;
    int*   txmap = (int*)d_ws;

    // 1) zero attn (64 MB)
#ifdef HAVE_TDM
    zero_attn_tdm_kernel<<<TDM_NBLOCKS, 256, 0, stream>>>(attn);
#else
    const int n4 = (B_ * TX * TY) / 4;
    zero_attn_kernel<<<4096, 256, 0, stream>>>((f32x4*)attn, n4);
#endif

    // 2) per-batch scan + inverse map
    scan_map_kernel<<<B_, TX, 0, stream>>>(durations, txmap);

    // 3) row gather into mu_y + one-hot scatter into attn
    gather_scatter_kernel<<<(B_ * TY) / 8, 256, 0, stream>>>(
        x, x_mask, y_mask, txmap, mu_y, attn);
}